// Net_69664369541652
// MI455X (gfx1250) — compile-verified
//
#include <hip/hip_runtime.h>
#include <math.h>

// Problem constants (match reference)
#define IDIM    80
#define ODIM    80
#define HDIM    512
#define KDIM    64      // top-k
#define NITER   8
#define NSHIFT  159     // selector shifts: 80 + 2*79 - 80 + 1
#define ESHIFT  81      // encoder shifts : 80 + 2*80 - 160 + 1
#define NBT     2048    // B*T
#define NTHREADS 256    // 8 wave32
#define MTILES  6       // 96 rows (81 valid encoder shifts)
#define NTILES  32      // 512 columns
#define KSTEPS  ((2*IDIM)/4)   // K = 160, 4 per V_WMMA_F32_16X16X4_F32

typedef __attribute__((ext_vector_type(2))) float v2f;
typedef __attribute__((ext_vector_type(8))) float v8f;

__global__ void init_ws_kernel(float* ws) {
    int i = threadIdx.x;
    if (i < 9) ws[i] = 0.0f;   // ws[0..7] = per-iter loss sums, ws[8] = valid count
}

__global__ void __launch_bounds__(NTHREADS)
mp_kernel(const float* __restrict__ x, const float* __restrict__ y,
          const float* __restrict__ enc_w, const float* __restrict__ enc_b,
          const float* __restrict__ dec_w, const float* __restrict__ dec_b,
          float* __restrict__ ws)
{
    __shared__ float s_xres[ODIM], s_yres[ODIM], s_yorig[ODIM], s_yatt[ODIM];
    __shared__ float s_xpad[256];       // selector padded residual (238 used)
    __shared__ float s_ypad[256];       // encoder padded y_att    (240 used)
    __shared__ float s_sim[NSHIFT];
    __shared__ float s_e[96];           // shift energies (81 used, tiles run to 96)
    __shared__ float s_h[HDIM], s_g[HDIM], s_mcur[HDIM], s_hmask[HDIM];
    __shared__ float s_xext[2*ODIM];
    __shared__ float s_red[NTHREADS];
    __shared__ int   s_idx[NTHREADS];
    __shared__ int   s_theta, s_hind;
    __shared__ float s_scal[4];

    const int tid = threadIdx.x;
    const int bt  = blockIdx.x;
    const float* xin = x + (size_t)bt * IDIM;
    const float* yin = y + (size_t)bt * ODIM;

    // Warm the weight streams into cache (global_prefetch_b8); ~128B granularity.
    for (int off = tid*32; off < 2*IDIM*HDIM; off += NTHREADS*32) {
        __builtin_prefetch(enc_w + off, 0, 3);
        __builtin_prefetch(dec_w + off, 0, 3);
    }

    // ---- per-(b,t) state init ----
    for (int i = tid; i < ODIM; i += NTHREADS) {
        s_xres[i] = xin[i]; s_yres[i] = yin[i]; s_yorig[i] = yin[i];
    }
    for (int j = tid; j < HDIM; j += NTHREADS) s_hmask[j] = 0.0f;

    // global (~seq_mask).sum(): count of y != 0 (constant over iterations)
    {
        s_red[tid] = (tid < ODIM) ? ((yin[tid] != 0.0f) ? 1.0f : 0.0f) : 0.0f;
        __syncthreads();
        for (int off = NTHREADS/2; off > 0; off >>= 1) {
            if (tid < off) s_red[tid] += s_red[tid+off];
            __syncthreads();
        }
        if (tid == 0) atomicAdd(&ws[8], s_red[0]);
    }
    __syncthreads();

    for (int iter = 0; iter < NITER; ++iter) {
        // ================= selector =================
        for (int p = tid; p < 256; p += NTHREADS)
            s_xpad[p] = (p >= ODIM-1 && p < 2*ODIM-1) ? s_xres[p-(ODIM-1)] : 0.0f;
        __syncthreads();
        if (tid == 0) {
            float ny = 0.0f;
            for (int i = 0; i < ODIM; ++i) ny += s_yres[i]*s_yres[i];
            s_scal[0] = sqrtf(ny);
        }
        __syncthreads();
        const float ny = s_scal[0];
        for (int s = tid; s < NSHIFT; s += NTHREADS) {
            float num = 0.0f, w2 = 0.0f;
            for (int i = 0; i < ODIM; ++i) {
                float v = s_xpad[s+i];
                num += s_yres[i]*v; w2 += v*v;
            }
            float denom = ny * sqrtf(w2);
            s_sim[s] = (denom == 0.0f) ? 0.0f : num/denom;
        }
        __syncthreads();
        if (tid == 0) {  // first-max tie-break like jnp.argmax
            float best = s_sim[0]; int bi = 0;
            for (int s = 1; s < NSHIFT; ++s) if (s_sim[s] > best) { best = s_sim[s]; bi = s; }
            s_theta = bi;
        }
        __syncthreads();
        const int theta = s_theta;

        // attn = softmax(y_align_opt * y); y_att = y_align_opt * attn
        if (tid < ODIM) s_red[tid] = s_xpad[theta+tid] * s_yorig[tid];
        __syncthreads();
        if (tid == 0) {
            float m = s_red[0];
            for (int i = 1; i < ODIM; ++i) m = fmaxf(m, s_red[i]);
            float ssum = 0.0f;
            for (int i = 0; i < ODIM; ++i) ssum += __expf(s_red[i]-m);
            s_scal[1] = m; s_scal[2] = ssum;
        }
        __syncthreads();
        if (tid < ODIM) {
            float attn = __expf(s_red[tid]-s_scal[1]) / s_scal[2];
            s_yatt[tid] = s_xpad[theta+tid] * attn;
        }
        __syncthreads();

        // x_res -= x_ele  (reverse shift: x_ele[i] = y_att[i + 79 - theta] if in range)
        if (tid < IDIM) {
            int q = tid + (ODIM-1) - theta;
            float xe = (q >= 0 && q < ODIM) ? s_yatt[q] : 0.0f;
            s_xres[tid] -= xe;
        }
        // encoder padded input + clear energies
        for (int p = tid; p < 256; p += NTHREADS)
            s_ypad[p] = (p >= IDIM && p < 2*IDIM) ? s_yatt[p-IDIM] : 0.0f;
        for (int s = tid; s < 96; s += NTHREADS) s_e[s] = 0.0f;
        __syncthreads();

        // ========== encoder energies via V_WMMA_F32_16X16X4_F32 ==========
        // h[s,j] = enc_b[j] + sum_k ypad[s+k]*enc_w[k,j];  e[s] = sum_j h[s,j]^2
        // Each wave owns one ntile (4 per wave); B fragments live in registers
        // and are reused across 6 independent C accumulators (one per mtile),
        // giving a 6-deep independent WMMA chain per k-step.
        {
            const int lane = tid & 31;
            const int wave = tid >> 5;
            const int half = lane >> 4;     // A/B K-striping: lanes 0-15 -> K{0,1}, 16-31 -> K{2,3}
            const int l15  = lane & 15;
            for (int nt = 0; nt < NTILES/8; ++nt) {
                const int ntile = wave + nt*8;
                const int col   = ntile*16 + l15;     // B/C column
                v8f c[MTILES];
                const float bias = enc_b[col];
                #pragma unroll
                for (int m = 0; m < MTILES; ++m)
                    #pragma unroll
                    for (int r = 0; r < 8; ++r) c[m][r] = bias;
                for (int kk = 0; kk < KSTEPS; ++kk) {
                    const int kb = kk*4 + half*2;
                    v2f b;
                    b.x = enc_w[(size_t)(kb    ) * HDIM + col];
                    b.y = enc_w[(size_t)(kb + 1) * HDIM + col];
                    #pragma unroll
                    for (int m = 0; m < MTILES; ++m) {
                        v2f a;
                        a.x = s_ypad[m*16 + l15 + kb];
                        a.y = s_ypad[m*16 + l15 + kb + 1];
                        c[m] = __builtin_amdgcn_wmma_f32_16x16x4_f32(
                                   false, a, false, b, (short)0, c[m], false, false);
                    }
                }
                // per row s: reduce c[m][r]^2 across the 16 lanes sharing that row
                #pragma unroll
                for (int m = 0; m < MTILES; ++m) {
                    #pragma unroll
                    for (int r = 0; r < 8; ++r) {
                        float v = c[m][r]*c[m][r];
                        v += __shfl_xor(v, 1, 32);
                        v += __shfl_xor(v, 2, 32);
                        v += __shfl_xor(v, 4, 32);
                        v += __shfl_xor(v, 8, 32);
                        int srow = m*16 + r + half*8;   // C/D layout: lanes>=16 hold M=r+8
                        if (l15 == 0 && srow < ESHIFT) atomicAdd(&s_e[srow], v);
                    }
                }
            }
        }
        __syncthreads();
        if (tid == 0) {
            float best = s_e[0]; int bi = 0;
            for (int s = 1; s < ESHIFT; ++s) if (s_e[s] > best) { best = s_e[s]; bi = s; }
            s_hind = bi;
        }
        __syncthreads();
        const int hind = s_hind;

        // recompute winning h row densely (512 x K160 GEMV), apply previous mask
        for (int j = tid; j < HDIM; j += NTHREADS) {
            float acc = enc_b[j];
            for (int k = 0; k < 2*IDIM; ++k)
                acc += s_ypad[hind + k] * enc_w[(size_t)k*HDIM + j];
            if (iter > 0 && s_hmask[j] > 0.0f) acc = 0.0f;
            s_h[j]    = acc;
            s_g[j]    = acc*acc;
            s_mcur[j] = 0.0f;
        }
        __syncthreads();

        // top-64 of g (jax top_k tie-break: earlier index wins)
        for (int round = 0; round < KDIM; ++round) {
            float bv = -1.0f; int bj = 0x7fffffff;
            for (int j = tid; j < HDIM; j += NTHREADS) {
                float v = s_g[j];
                if (v > bv || (v == bv && j < bj)) { bv = v; bj = j; }
            }
            s_red[tid] = bv; s_idx[tid] = bj;
            __syncthreads();
            for (int off = NTHREADS/2; off > 0; off >>= 1) {
                if (tid < off) {
                    float ov = s_red[tid+off]; int oj = s_idx[tid+off];
                    if (ov > s_red[tid] || (ov == s_red[tid] && oj < s_idx[tid])) {
                        s_red[tid] = ov; s_idx[tid] = oj;
                    }
                }
                __syncthreads();
            }
            if (tid == 0) {
                int w = s_idx[0];
                s_mcur[w] = 1.0f;
                s_g[w]    = -1.0f;
            }
            __syncthreads();
        }
        for (int j = tid; j < HDIM; j += NTHREADS) {
            s_hmask[j] += s_mcur[j];
            if (!(s_mcur[j] > 0.0f)) s_h[j] = 0.0f;
        }
        __syncthreads();

        // decoder: x_ext = h @ dec_w + dec_b   (160 outputs)
        for (int d = tid; d < 2*ODIM; d += NTHREADS) {
            float acc = dec_b[d];
            for (int j = 0; j < HDIM; ++j)
                acc += s_h[j] * dec_w[(size_t)j*(2*ODIM) + d];
            s_xext[d] = acc;
        }
        __syncthreads();

        // y_ele gather, masked MSE contribution, residual update
        float ls = 0.0f;
        if (tid < ODIM) {
            int idx = IDIM + tid - hind;        // offs = 80 + i - h_ind, in [0,159]
            float ye = s_xext[idx];
            if (s_yorig[tid] != 0.0f) {
                float d = ye - s_yres[tid];
                ls = d*d;
            }
            s_yres[tid] -= ye;
        }
        s_red[tid] = ls;
        __syncthreads();
        for (int off = NTHREADS/2; off > 0; off >>= 1) {
            if (tid < off) s_red[tid] += s_red[tid+off];
            __syncthreads();
        }
        if (tid == 0) atomicAdd(&ws[iter], s_red[0]);
        __syncthreads();
    }
}

__global__ void finalize_kernel(const float* __restrict__ ws, float* __restrict__ out) {
    if (threadIdx.x == 0 && blockIdx.x == 0) {
        float cnt = ws[8];
        if (cnt < 1.0f) cnt = 1.0f;
        float s = 0.0f;
        for (int i = 0; i < NITER; ++i) s += ws[i] / cnt;
        out[0] = s / (float)NITER;
    }
}

extern "C" void kernel_launch(void* const* d_in, const int* in_sizes, int n_in,
                              void* d_out, int out_size, void* d_ws, size_t ws_size,
                              hipStream_t stream) {
    const float* x     = (const float*)d_in[0];
    const float* y     = (const float*)d_in[1];
    const float* enc_w = (const float*)d_in[2];
    const float* enc_b = (const float*)d_in[3];
    const float* dec_w = (const float*)d_in[4];
    const float* dec_b = (const float*)d_in[5];
    float* out = (float*)d_out;
    float* ws  = (float*)d_ws;

    init_ws_kernel<<<1, 32, 0, stream>>>(ws);
    mp_kernel<<<NBT, NTHREADS, 0, stream>>>(x, y, enc_w, enc_b, dec_w, dec_b, ws);
    finalize_kernel<<<1, 32, 0, stream>>>(ws, out);
}